// KNNMutationSite_24859270709372
// MI455X (gfx1250) — compile-verified
//
#include <hip/hip_runtime.h>

#define CA_ID 1

typedef __attribute__((ext_vector_type(2))) float v2f;
typedef __attribute__((ext_vector_type(8))) float v8f;

// ---------------------------------------------------------------------------
// Kernel 1: single wave32. Deterministically extract centers (mut_ca nodes in
// ascending index order), their squared norms, graph ids (-1 == invalid slot);
// count of mut_ca nodes; and per-graph segment starts (batch is sorted).
// ---------------------------------------------------------------------------
__global__ __launch_bounds__(32) void knn_prep(
    const float* __restrict__ pos, const int* __restrict__ atom,
    const unsigned char* __restrict__ ismut, const int* __restrict__ batch,
    int N, const int* __restrict__ d_numc,
    float4* __restrict__ cpos, float* __restrict__ c2,
    int* __restrict__ cgraph,
    int* __restrict__ cnt, int* __restrict__ gstart)
{
    const int lane = threadIdx.x;
    int numc = *d_numc; if (numc > 256) numc = 256;

    // init 256 center slots: cgraph = -1 marks invalid (batch ids are >= 0)
    for (int r = lane; r < 256; r += 32) {
        cpos[r]   = make_float4(0.f, 0.f, 0.f, 0.f);
        c2[r]     = 0.f;
        cgraph[r] = -1;
    }

    // per-graph segment start: lower_bound of g in sorted batch
    {
        int g = lane;
        int lo = 0, hi = N;
        while (lo < hi) {
            int mid = (lo + hi) >> 1;
            if (batch[mid] < g) lo = mid + 1; else hi = mid;
        }
        gstart[g] = lo;
        if (lane == 0) gstart[32] = N;
    }
    __syncthreads();

    // ballot/popcount scan: rank = #mut_ca with smaller index (deterministic)
    int count = 0;
    for (int base = 0; base < N; base += 32) {
        int i = base + lane;
        bool m = (i < N) && (ismut[i] != 0) && (atom[i] == CA_ID);
        unsigned int bm = (unsigned int)__ballot(m);   // wave32: low 32 bits
        int rank = count + __popc(bm & ((1u << lane) - 1u));
        if (m && rank < numc) {
            float x = pos[3*i + 0], y = pos[3*i + 1], z = pos[3*i + 2];
            cpos[rank]   = make_float4(x, y, z, 0.f);
            c2[rank]     = x*x + y*y + z*z;
            cgraph[rank] = batch[i];
        }
        count += __popc(bm);
    }
    if (lane == 0) cnt[0] = count;
}

// ---------------------------------------------------------------------------
// Kernel 2: one wave32 per 16-node tile. Cross term node·center for
// 16 nodes x 256 centers via 16x V_WMMA_F32_16X16X4_F32 (K=4, padded 4th
// coord = 0), then d2 = x2 + c2 - 2*cross, clamp >= 0, same-graph mask as a
// branchless v_cndmask to +inf, min-reduce over centers, write per-node min
// distance (0 for mut_ca nodes).
//
// A layout (32-bit 16x4): lanes 0-15 row M=lane K=0,1 ; lanes 16-31 K=2,3.
// B layout (4x16) mirrors with N across lanes -> loads identically from the
// center rows (x,y,z,0). C/D: VGPR r, lane L -> (M = r + 8*(L>=16), N = L&15).
// ---------------------------------------------------------------------------
__global__ __launch_bounds__(32) void knn_dist(
    const float* __restrict__ pos, const int* __restrict__ atom,
    const unsigned char* __restrict__ ismut, const int* __restrict__ batch,
    int N,
    const float4* __restrict__ cpos, const float* __restrict__ c2,
    const int* __restrict__ cgraph,
    float* __restrict__ dist)
{
    const int lane   = threadIdx.x;
    const int lo     = lane & 15;
    const int hi     = lane >> 4;
    const int node16 = blockIdx.x * 16;

    __shared__ float sx2[16];
    __shared__ int   sb[16];

    int na = node16 + lo; if (na > N - 1) na = N - 1;   // clamp, keep EXEC full
    float ax = pos[3*na + 0], ay = pos[3*na + 1], az = pos[3*na + 2];
    if (hi == 0) { sx2[lo] = ax*ax + ay*ay + az*az; sb[lo] = batch[na]; }
    __syncthreads();

    v2f A;
    A.x = hi ? az  : ax;
    A.y = hi ? 0.f : ay;

    float x2r[8]; int br[8];
#pragma unroll
    for (int r = 0; r < 8; ++r) { x2r[r] = sx2[r + 8*hi]; br[r] = sb[r + 8*hi]; }

    const float INF = __builtin_inff();
    float dmin[8];
#pragma unroll
    for (int r = 0; r < 8; ++r) dmin[r] = INF;

#pragma unroll
    for (int ct = 0; ct < 16; ++ct) {
        int c = ct * 16 + lo;
        float4 cp = cpos[c];
        v2f B;
        B.x = hi ? cp.z : cp.x;
        B.y = hi ? 0.f  : cp.y;

        v8f acc = {};
        acc = __builtin_amdgcn_wmma_f32_16x16x4_f32(
            /*neg_a=*/false, A, /*neg_b=*/false, B,
            /*c_mod=*/(short)0, acc, /*reuse_a=*/false, /*reuse_b=*/false);

        float cc2 = c2[c];
        int   cg  = cgraph[c];      // -1 for invalid slots => never matches
#pragma unroll
        for (int r = 0; r < 8; ++r) {
            float d2 = fmaxf(x2r[r] + cc2 - 2.0f * acc[r], 0.f);
            d2 = (br[r] == cg) ? d2 : INF;     // data select, not exec branch
            dmin[r] = fminf(dmin[r], d2);
        }
    }

    // min over the 16 lanes of each half (XOR shuffles stay within halves)
#pragma unroll
    for (int r = 0; r < 8; ++r) {
#pragma unroll
        for (int off = 8; off >= 1; off >>= 1)
            dmin[r] = fminf(dmin[r], __shfl_xor(dmin[r], off, 32));
    }

    if (lo == 0) {
#pragma unroll
        for (int r = 0; r < 8; ++r) {
            int n = node16 + 8*hi + r;
            if (n < N) {
                bool mc = (ismut[n] != 0) && (atom[n] == CA_ID);
                dist[n] = mc ? 0.f : dmin[r];
            }
        }
    }
}

// ---------------------------------------------------------------------------
// Kernel 3: one block per graph. Stable-rank selection: node i selected iff
// (#j in graph with d_j < d_i) + (#j with d_j == d_i && j < i) < k.
// Matches stable lexsort((dist, batch)) tie-breaking. Handles the
// "no mut_ca anywhere -> all true" fallback. dist[j] is a block-uniform
// address stream -> broadcast-friendly, stays in WGP$/L2.
// ---------------------------------------------------------------------------
__global__ __launch_bounds__(256) void knn_select(
    const float* __restrict__ dist, const int* __restrict__ gstart,
    const int* __restrict__ cnt, const int* __restrict__ d_k,
    float* __restrict__ out)
{
    const int g     = blockIdx.x;
    const int start = gstart[g];
    const int end   = gstart[g + 1];
    const int tid   = threadIdx.x;
    const int kk    = *d_k;

    if (cnt[0] == 0) {
        for (int i = start + tid; i < end; i += 256) out[i] = 1.0f;
        return;
    }

    for (int i = start + tid; i < end; i += 256) {
        float di = dist[i];
        int   c  = 0;
        for (int j = start; j < end; ++j) {
            float dj = dist[j];
            c += (dj < di) || ((dj == di) && (j < i));
        }
        out[i] = (c < kk) ? 1.0f : 0.0f;
    }
}

// ---------------------------------------------------------------------------
// Launch
// ---------------------------------------------------------------------------
extern "C" void kernel_launch(void* const* d_in, const int* in_sizes, int n_in,
                              void* d_out, int out_size, void* d_ws, size_t ws_size,
                              hipStream_t stream) {
    const float*         pos    = (const float*)d_in[0];
    const int*           atom   = (const int*)d_in[1];
    const unsigned char* ismut  = (const unsigned char*)d_in[2]; // bool, 1 byte
    const int*           batch  = (const int*)d_in[3];           // jnp demotes int64->int32
    const int*           d_numc = (const int*)d_in[4];
    const int*           d_k    = (const int*)d_in[5];
    float*               out    = (float*)d_out;

    const int N = in_sizes[1]; // atom_name_ids element count

    // workspace layout (d_ws is hipMalloc'd -> 256B aligned)
    char* w = (char*)d_ws;
    float4* cpos   = (float4*)(w + 0);       // 256 * 16 B = 4096
    float*  c2     = (float*) (w + 4096);    // 1024
    int*    cgraph = (int*)   (w + 5120);    // 1024
    int*    gstart = (int*)   (w + 6144);    // 33 * 4 = 132
    int*    cnt    = (int*)   (w + 6400);    // 4
    float*  dist   = (float*) (w + 8192);    // N * 4

    knn_prep<<<1, 32, 0, stream>>>(pos, atom, ismut, batch, N, d_numc,
                                   cpos, c2, cgraph, cnt, gstart);

    int tiles = (N + 15) / 16;
    knn_dist<<<tiles, 32, 0, stream>>>(pos, atom, ismut, batch, N,
                                       cpos, c2, cgraph, dist);

    knn_select<<<32, 256, 0, stream>>>(dist, gstart, cnt, d_k, out);
}